// JointLoss_80126909874420
// MI455X (gfx1250) — compile-verified
//
#include <hip/hip_runtime.h>

#define NROWS    8192
#define DDIM     128
#define RECON_N  (8192 * 784)
#define TAU_INV  10.0f

typedef __attribute__((ext_vector_type(2))) float v2f;
typedef __attribute__((ext_vector_type(8))) float v8f;

// ---------------------------------------------------------------- utilities

__global__ void zero_ws_kernel(float* ws, int n) {
    int i = blockIdx.x * blockDim.x + threadIdx.x;
    if (i < n) ws[i] = 0.0f;
}

__device__ __forceinline__ float blockReduceSum(float v) {
    __shared__ float smem[8];
    const int lane = threadIdx.x & 31;
    const int wave = threadIdx.x >> 5;
    #pragma unroll
    for (int o = 16; o; o >>= 1) v += __shfl_down(v, o, 32);
    if (lane == 0) smem[wave] = v;
    __syncthreads();
    const int nw = blockDim.x >> 5;
    v = (threadIdx.x < nw) ? smem[threadIdx.x] : 0.0f;
    if (wave == 0) {
        #pragma unroll
        for (int o = 4; o; o >>= 1) v += __shfl_down(v, o, 32);
    }
    return v;  // valid in thread 0
}

// ------------------------------------------------- recon_loss (bandwidth-bound)

__global__ __launch_bounds__(256) void recon_kernel(const float4* __restrict__ x,
                                                    const float4* __restrict__ y,
                                                    float* __restrict__ acc, int n4) {
    float s = 0.0f;
    for (int i = blockIdx.x * blockDim.x + threadIdx.x; i < n4;
         i += gridDim.x * blockDim.x) {
        float4 a = x[i], b = y[i];
        float dx = a.x - b.x, dy = a.y - b.y, dz = a.z - b.z, dw = a.w - b.w;
        s += dx * dx + dy * dy + dz * dz + dw * dw;
    }
    s = blockReduceSum(s);
    if (threadIdx.x == 0) atomicAdd(acc, s);
}

// ------------------------------------------------- dist_loss: 4*sum(a^2) - (sum a)^2

__global__ __launch_bounds__(256) void dist_kernel(const float* __restrict__ P,
                                                   float* __restrict__ acc) {
    const int idx = blockIdx.x * blockDim.x + threadIdx.x;   // 0 .. 2048*128-1
    const int g = idx >> 7;
    const int d = idx & 127;
    const float* base = P + g * 4 * DDIM + d;
    float a0 = base[0], a1 = base[DDIM], a2 = base[2 * DDIM], a3 = base[3 * DDIM];
    float sum = a0 + a1 + a2 + a3;
    float sq  = a0 * a0 + a1 * a1 + a2 * a2 + a3 * a3;
    float v = 4.0f * sq - sum * sum;
    v = blockReduceSum(v);
    if (threadIdx.x == 0) atomicAdd(acc, v);
}

// ------------------------------------------------- closs main: fused WMMA GEMM + exp + masked row sums
// grid = (64, 8), block = 256 (8 waves). Wave w of block bx owns row tile bx*8+w
// (16 rows); blockIdx.y selects a 1024-column chunk (64 tiles of 16 cols).

__global__ __launch_bounds__(256) void closs_main_kernel(const float* __restrict__ P,
                                                         float* __restrict__ pos,
                                                         float* __restrict__ neg) {
    const int lane = threadIdx.x & 31;
    const int wave = threadIdx.x >> 5;
    const int half = lane >> 4;          // 0: K{0,1}, rows M+0..7 of C | 1: K{2,3}, rows M+8..15
    const int l15  = lane & 15;
    const int rowBase  = (blockIdx.x * 8 + wave) * 16;
    const int colBase0 = blockIdx.y * 1024;

    // A fragments: 16x128 strip, 32 K-slices of 16x4. ISA layout: lanes 0-15 hold
    // (M=lane, K=4s+{0,1}), lanes 16-31 hold (M=lane-16, K=4s+{2,3}) -> one float2 load.
    v2f a[32];
    {
        const float* ap = P + (rowBase + l15) * DDIM + 2 * half;
        #pragma unroll
        for (int s = 0; s < 32; ++s) a[s] = *(const v2f*)(ap + 4 * s);
    }

    float posr[8], negr[8];
    #pragma unroll
    for (int r = 0; r < 8; ++r) { posr[r] = 0.0f; negr[r] = 0.0f; }

    for (int ct = 0; ct < 64; ++ct) {
        const int colBase = colBase0 + ct * 16;
        // B 4x16 slice: lane holds (N=l15, K=4s + 2*half + {0,1}) -> same float2 pattern.
        const float* bp = P + (colBase + l15) * DDIM + 2 * half;
        __builtin_prefetch(bp + 16 * DDIM, 0, 1);   // next column tile -> global_prefetch_b8

        v8f c = {};
        #pragma unroll
        for (int s = 0; s < 32; ++s) {
            v2f b = *(const v2f*)(bp + 4 * s);
            c = __builtin_amdgcn_wmma_f32_16x16x4_f32(
                    /*neg_a=*/false, a[s], /*neg_b=*/false, b,
                    /*c_mod=*/(short)0, c, /*reuse_a=*/false, /*reuse_b=*/false);
        }

        // C layout: VGPR r -> row rowBase + 8*half + r, column colBase + l15.
        const int jcol = colBase + l15;
        const int gj   = jcol >> 2;                 // group id of column
        #pragma unroll
        for (int r = 0; r < 8; ++r) {
            const int irow = rowBase + (half << 3) + r;
            float sim = c[r] * TAU_INV;
            float e = __expf(sim);
            bool samegroup = ((irow >> 2) == gj);
            posr[r] += (samegroup && (sim != 1.0f)) ? e : 0.0f;
            negr[r] += samegroup ? 0.0f : e;
        }
    }

    // Reduce each row sum across its 16-lane half, then accumulate globally.
    #pragma unroll
    for (int r = 0; r < 8; ++r) {
        float p = posr[r], n = negr[r];
        #pragma unroll
        for (int o = 8; o; o >>= 1) {
            p += __shfl_down(p, o, 16);
            n += __shfl_down(n, o, 16);
        }
        if (l15 == 0) {
            const int irow = rowBase + (half << 3) + r;
            atomicAdd(&pos[irow], p);
            atomicAdd(&neg[irow], n);
        }
    }
}

// ------------------------------------------------- closs row pass: -log(pos/(pos+neg))

__global__ __launch_bounds__(256) void closs_rows_kernel(const float* __restrict__ pos,
                                                         const float* __restrict__ neg,
                                                         float* __restrict__ acc) {
    const int i = blockIdx.x * blockDim.x + threadIdx.x;
    float p = pos[i], n = neg[i];
    float c = -__logf(p / (p + n));
    c = blockReduceSum(c);
    if (threadIdx.x == 0) atomicAdd(acc, c);
}

// ------------------------------------------------- finalize: write (loss, closs, recon, dist)

__global__ void finalize_kernel(const float* __restrict__ ws, float* __restrict__ out) {
    float recon = ws[0] / (float)RECON_N;
    float dist  = ws[1] / (2048.0f * 6.0f * 128.0f);
    float closs = ws[2] / (float)NROWS;
    out[0] = recon + closs + dist;
    out[1] = closs;
    out[2] = recon;
    out[3] = dist;
}

// ---------------------------------------------------------------- launch

extern "C" void kernel_launch(void* const* d_in, const int* in_sizes, int n_in,
                              void* d_out, int out_size, void* d_ws, size_t ws_size,
                              hipStream_t stream) {
    (void)in_sizes; (void)n_in; (void)out_size; (void)ws_size;
    const float* P  = (const float*)d_in[0];   // projections 8192x128
    const float* xr = (const float*)d_in[1];   // xrecon      8192x784
    const float* rl = (const float*)d_in[2];   // recon_label 8192x784
    float* out = (float*)d_out;
    float* ws  = (float*)d_ws;                 // [0]=recon [1]=dist [2]=closs [3]=pad
    float* pos = ws + 4;                       // 8192 floats
    float* neg = ws + 4 + NROWS;               // 8192 floats

    const int zn = 4 + 2 * NROWS;
    zero_ws_kernel<<<(zn + 255) / 256, 256, 0, stream>>>(ws, zn);

    recon_kernel<<<512, 256, 0, stream>>>((const float4*)xr, (const float4*)rl,
                                          ws + 0, RECON_N / 4);
    dist_kernel<<<(NROWS / 4) * DDIM / 256, 256, 0, stream>>>(P, ws + 1);

    dim3 grid(64, 8);  // 512 row tiles x 8 column chunks
    closs_main_kernel<<<grid, 256, 0, stream>>>(P, pos, neg);
    closs_rows_kernel<<<NROWS / 256, 256, 0, stream>>>(pos, neg, ws + 2);

    finalize_kernel<<<1, 1, 0, stream>>>(ws, out);
}